// GraphConvolutionLayer_DGCNN_23605140259236
// MI455X (gfx1250) — compile-verified
//
#include <hip/hip_runtime.h>
#include <math.h>

typedef __attribute__((ext_vector_type(2))) float v2f;
typedef __attribute__((ext_vector_type(8))) float v8f;

#define DIM 128

// ---------------------------------------------------------------------------
// Phase 2: pooled[dst] += X[src]  (pooled pre-initialized to X => (A+I)X)
// One wave32 per edge; lane l handles columns [4l, 4l+4).
// ---------------------------------------------------------------------------
__global__ __launch_bounds__(256)
void dgcnn_scatter_add(const float* __restrict__ x,
                       const int* __restrict__ src,
                       const int* __restrict__ dst,
                       float* __restrict__ pooled, int E)
{
    int edge = (int)((blockIdx.x * blockDim.x + threadIdx.x) >> 5);
    int lane = threadIdx.x & 31;
    if (edge >= E) return;
    int s = src[edge];
    int d = dst[edge];
    const float4 v = ((const float4*)(x + (size_t)s * DIM))[lane];
    float* o = pooled + (size_t)d * DIM + lane * 4;
    atomicAdd(o + 0, v.x);
    atomicAdd(o + 1, v.y);
    atomicAdd(o + 2, v.z);
    atomicAdd(o + 3, v.w);
}

// ---------------------------------------------------------------------------
// Phase 3: out = tanh((pooled @ W + b) / deg)
// fp32 WMMA 16x16x4. 8 waves/block, 1 row-tile (16 rows) per wave, full 128
// output columns held in 8 v8f accumulators. W staged in LDS, pre-swizzled to
// B-fragment order so each B fragment is one ds_load_b64.
//
// Fragment layouts (ISA 7.12.2, 32-bit, wave32):
//   A 16x4 : lane l -> row M = l&15 ; VGPR0 = K0+2*(l>>4), VGPR1 = K0+2*(l>>4)+1
//   B 4x16 : lane l -> col N = l&15 ; VGPR0 = row K0+2*(l>>4), VGPR1 = +1
//   C 16x16: VGPR j : lanes 0-15 -> M=j, lanes 16-31 -> M=j+8 ; N = l&15
// ---------------------------------------------------------------------------
__global__ __launch_bounds__(256)
void dgcnn_wmma_gemm(const float* __restrict__ pooled,
                     const float* __restrict__ W,
                     const float* __restrict__ bias,
                     const float* __restrict__ deg,
                     float* __restrict__ out,
                     int nRows, int nTiles)
{
    // lw[kt][nt][lane] : kt = K/4 step (32), nt = N/16 tile (8), lane (32). 64 KB.
    __shared__ v2f lw[32 * 8 * 32];

    // Cooperative swizzled fill of W into LDS.
    for (int i = threadIdx.x; i < 32 * 8 * 32; i += blockDim.x) {
        int kt = i >> 8;            // i / 256
        int nt = (i >> 5) & 7;
        int l  = i & 31;
        int k0  = kt * 4 + ((l >> 4) << 1);
        int col = nt * 16 + (l & 15);
        v2f v;
        v.x = W[(k0 + 0) * DIM + col];
        v.y = W[(k0 + 1) * DIM + col];
        lw[i] = v;
    }
    __syncthreads();

    int waveId = threadIdx.x >> 5;
    int lane   = threadIdx.x & 31;
    int tile   = blockIdx.x * 8 + waveId;
    if (tile >= nTiles) return;          // wave-uniform: EXEC stays all-1s for WMMA

    int rowBase = tile * 16;
    int half    = lane >> 4;             // 0 or 1
    int rsub    = lane & 15;

    v8f acc[8];
    #pragma unroll
    for (int nt = 0; nt < 8; ++nt) acc[nt] = (v8f)(0.0f);

    // A fragment source: row (rowBase+rsub), starting column half*2; the two
    // K values per VGPR pair are adjacent in memory -> single 8-byte load.
    int arow = rowBase + rsub;
    if (arow >= nRows) arow = nRows - 1;       // safe clamp (N % 16 == 0 here)
    const float* aPtr = pooled + (size_t)arow * DIM + half * 2;

    for (int kt = 0; kt < 32; ++kt) {
        v2f a = *(const v2f*)(aPtr + kt * 4);
        #pragma unroll
        for (int nt = 0; nt < 8; ++nt) {
            v2f b = lw[(kt << 8) | (nt << 5) | lane];
            acc[nt] = __builtin_amdgcn_wmma_f32_16x16x4_f32(
                false, a, false, b, (short)0, acc[nt], false, false);
        }
    }

    // Epilogue: +bias, * (1/deg), tanh, store.
    float rd[8];
    #pragma unroll
    for (int j = 0; j < 8; ++j) {
        int row = rowBase + j + half * 8;
        rd[j] = (row < nRows) ? (1.0f / deg[row]) : 0.0f;
    }
    #pragma unroll
    for (int nt = 0; nt < 8; ++nt) {
        int col = nt * 16 + rsub;
        float bb = bias[col];
        #pragma unroll
        for (int j = 0; j < 8; ++j) {
            int row = rowBase + j + half * 8;
            if (row < nRows) {
                float v = (acc[nt][j] + bb) * rd[j];
                out[(size_t)row * DIM + col] = tanhf(v);
            }
        }
    }
}

// ---------------------------------------------------------------------------
// Inputs (setup_inputs order):
//   d_in[0] input_tensor  float32 [N,128]
//   d_in[1] edge_index    int32   [2,E]   (row 0 = src, row 1 = dst)
//   d_in[2] node_degree   float32 [N,1]
//   d_in[3] W             float32 [128,128]
//   d_in[4] b             float32 [128]
// d_out: float32 [N,128]
// d_ws : >= N*128*4 bytes used for pooled accumulator
// ---------------------------------------------------------------------------
extern "C" void kernel_launch(void* const* d_in, const int* in_sizes, int n_in,
                              void* d_out, int out_size, void* d_ws, size_t ws_size,
                              hipStream_t stream) {
    const float* x    = (const float*)d_in[0];
    const int*   ei   = (const int*)d_in[1];
    const float* degm = (const float*)d_in[2];
    const float* W    = (const float*)d_in[3];
    const float* bias = (const float*)d_in[4];
    float*       out  = (float*)d_out;
    float*       pooled = (float*)d_ws;

    const int N = in_sizes[0] / DIM;
    const int E = in_sizes[1] / 2;

    // pooled = X  (the "+ I" residual term)
    hipMemcpyAsync(pooled, x, (size_t)N * DIM * sizeof(float),
                   hipMemcpyDeviceToDevice, stream);

    // pooled += scatter_add(X[src] -> dst)   (1 wave per edge, 8 edges/block)
    int edgeBlocks = (E + 7) / 8;
    dgcnn_scatter_add<<<edgeBlocks, 256, 0, stream>>>(x, ei, ei + E, pooled, E);

    // out = tanh((pooled @ W + b) / deg)     (fp32 WMMA)
    int nTiles = (N + 15) / 16;
    int gemmBlocks = (nTiles + 7) / 8;
    dgcnn_wmma_gemm<<<gemmBlocks, 256, 0, stream>>>(pooled, W, bias, degm, out,
                                                    N, nTiles);
}